// ClipperEventEncoder_979252543758
// MI455X (gfx1250) — compile-verified
//
#include <hip/hip_runtime.h>

// CDNA5 (gfx1250) fused LIF + conv3x3(1->4) + ReLU + conv3x3(4->1) + ReLU + time-mean.
// One wave32 per 16x16 output tile; membrane state v in registers for the 20x20
// halo region across all 96 steps; convs exact-f32 via V_WMMA_F32_16X16X4_F32;
// interior tiles stage x[t] into LDS with the Tensor Data Mover (double-buffered,
// s_wait_tensorcnt), border tiles use clamped global loads.

typedef __attribute__((ext_vector_type(2))) float v2f;
typedef __attribute__((ext_vector_type(8))) float v8f;
typedef __attribute__((ext_vector_type(4))) unsigned int u32x4;
typedef __attribute__((ext_vector_type(8))) int i32x8;
typedef __attribute__((ext_vector_type(4))) int i32x4;

#define T_STEPS 96
#define IMG     512
#define TILE    16
#define SDIM    20   // TILE + 4 (spike/LIF region, halo 2)
#define HDIM    18   // TILE + 2 (conv1 output region, halo 1)
#define SN      (SDIM * SDIM)        // 400
#define HN      (HDIM * HDIM)        // 324
#define SPAD    (SN + 24)            // dump word at SN, masked tail reads stay in-array

__device__ __forceinline__ int iclamp(int x, int lo, int hi) {
    return x < lo ? lo : (x > hi ? hi : x);
}

// Issue a TDM load of a 20x20 f32 tile (row stride IMG) from gsrc into LDS at
// byte offset lds_off.  D# layout per CDNA5 ISA.
__device__ __forceinline__ void tdm_load_tile(const float* gsrc, unsigned lds_off) {
    unsigned long long ga = (unsigned long long)(uintptr_t)gsrc;
    // group0: count=1 | lds_addr | global_addr[56:0] | type=2
    unsigned g0w0 = __builtin_amdgcn_readfirstlane(1u);
    unsigned g0w1 = __builtin_amdgcn_readfirstlane(lds_off);
    unsigned g0w2 = __builtin_amdgcn_readfirstlane((unsigned)(ga & 0xFFFFFFFFu));
    unsigned g0w3 = __builtin_amdgcn_readfirstlane(
                        (unsigned)((ga >> 32) & 0x01FFFFFFu) | (2u << 30));
    // group1: data_size=4B; tensor 512x512; tile 20x20; dim0 stride 512
    int g1w0 = __builtin_amdgcn_readfirstlane((int)(2u << 16));            // data_size=2 (4B)
    int g1w1 = __builtin_amdgcn_readfirstlane((int)((unsigned)IMG << 16)); // tensor_dim0 lo
    int g1w2 = __builtin_amdgcn_readfirstlane((int)((unsigned)IMG << 16)); // dim0 hi | tensor_dim1 lo
    int g1w3 = __builtin_amdgcn_readfirstlane((int)((unsigned)SDIM << 16));// dim1 hi | tile_dim0=20
    int g1w4 = __builtin_amdgcn_readfirstlane((int)SDIM);                  // tile_dim1=20 | tile_dim2=0
    int g1w5 = __builtin_amdgcn_readfirstlane((int)IMG);                   // tensor_dim0_stride lo
    int g1w6 = __builtin_amdgcn_readfirstlane(0);
    int g1w7 = __builtin_amdgcn_readfirstlane(0);
    u32x4 g0 = {g0w0, g0w1, g0w2, g0w3};
    i32x8 g1 = {g1w0, g1w1, g1w2, g1w3, g1w4, g1w5, g1w6, g1w7};
    i32x4 gz4 = {0, 0, 0, 0};
    i32x8 gz8 = {0, 0, 0, 0, 0, 0, 0, 0};
    __builtin_amdgcn_tensor_load_to_lds(g0, g1, gz4, gz4, gz8, 0);
}

template <bool INTERIOR>
__device__ __forceinline__ void run_tile(const float* __restrict__ x_seq,
                                         const float* __restrict__ w1,
                                         const float* __restrict__ w2,
                                         float* __restrict__ out,
                                         float* sLDS, float* hLDS, float* yacc,
                                         float (*xbuf)[SN],
                                         int lane, int ox0, int oy0) {
    const int n  = lane & 15;                 // matrix col (N) / A row (M)
    const int kk = (lane >> 4) << 1;          // K pair base: 0 or 2

    // ---- weight fragments (B operands) ----
    v2f b1[3];
#pragma unroll
    for (int kb = 0; kb < 3; ++kb) {
        float e0 = 0.f, e1 = 0.f;
        int k0 = kb * 4 + kk;
        if (n < 4) {
            if (k0     < 9) e0 = w1[n * 9 + k0];
            if (k0 + 1 < 9) e1 = w1[n * 9 + k0 + 1];
        }
        b1[kb].x = e0; b1[kb].y = e1;
    }
    v2f b2[9];
#pragma unroll
    for (int tap = 0; tap < 9; ++tap) {
        float e0 = 0.f, e1 = 0.f;
        if (n == 0) { e0 = w2[kk * 9 + tap]; e1 = w2[(kk + 1) * 9 + tap]; }
        b2[tap].x = e0; b2[tap].y = e1;
    }

    // ---- per-lane loop-invariant conv offsets ----
    int  toff[6];      // conv1 tap word offsets in sLDS: dr*SDIM + dc
    bool kval[6];      // k < 9
#pragma unroll
    for (int kb = 0; kb < 3; ++kb)
#pragma unroll
        for (int tt = 0; tt < 2; ++tt) {
            int k   = kb * 4 + kk + tt;
            int kcl = (k < 9) ? k : 0;
            int dr = kcl / 3, dc = kcl - dr * 3;
            toff[kb * 2 + tt] = dr * SDIM + dc;
            kval[kb * 2 + tt] = (k < 9);
        }
    int c2[9];         // conv2 tap word offsets in hLDS (per lane)
#pragma unroll
    for (int tap = 0; tap < 9; ++tap) {
        int dr = tap / 3, dc = tap - dr * 3;
        c2[tap] = (dr * HDIM + n + dc) * 4 + kk;
    }

#pragma unroll
    for (int i = 0; i < 8; ++i) yacc[i * 32 + lane] = 0.f;

    float v[13];
#pragma unroll
    for (int i = 0; i < 13; ++i) v[i] = 0.f;

    if (INTERIOR) {   // kick off frame 0 via TDM
        const float* g0src = x_seq + (size_t)(oy0 - 2) * IMG + (ox0 - 2);
        tdm_load_tile(g0src, (unsigned)(uintptr_t)&xbuf[0][0]);
    }

    for (int t = 0; t < T_STEPS; ++t) {
        // ---- LIF: v <- (v+x)/2 ; spike ; hard reset ; spikes -> LDS ----
        if (INTERIOR) {
            if (t + 1 < T_STEPS) {
                const float* gn = x_seq + (size_t)(t + 1) * IMG * IMG +
                                  (size_t)(oy0 - 2) * IMG + (ox0 - 2);
                tdm_load_tile(gn, (unsigned)(uintptr_t)&xbuf[(t + 1) & 1][0]);
                __builtin_amdgcn_s_wait_tensorcnt(1);
            } else {
                __builtin_amdgcn_s_wait_tensorcnt(0);
            }
            const float* xc = &xbuf[t & 1][0];
#pragma unroll
            for (int i = 0; i < 13; ++i) {
                int e  = i * 32 + lane;
                int ec = (e < SN) ? e : (SN - 1);
                float xv = xc[ec];
                float nv = (v[i] + xv) * 0.5f;
                float sp = (nv >= 1.0f) ? 1.0f : 0.0f;
                v[i] = nv * (1.0f - sp);
                sLDS[(e < SN) ? e : SN] = sp;          // tail lanes -> dump word
            }
        } else {
            const float* xt = x_seq + (size_t)t * IMG * IMG;
            if (t + 1 < T_STEPS)
                __builtin_prefetch(x_seq + (size_t)(t + 1) * IMG * IMG +
                                   (size_t)iclamp(oy0 - 2, 0, IMG - 1) * IMG + ox0, 0, 0);
#pragma unroll
            for (int i = 0; i < 13; ++i) {
                int e  = i * 32 + lane;
                int ec = (e < SN) ? e : (SN - 1);
                int sy = ec / SDIM, sx = ec - sy * SDIM;
                int gy = oy0 + sy - 2, gx = ox0 + sx - 2;
                bool inb = (e < SN) && (gy >= 0) && (gy < IMG) &&
                           (gx >= 0) && (gx < IMG);
                float xv = xt[(size_t)iclamp(gy, 0, IMG - 1) * IMG +
                              iclamp(gx, 0, IMG - 1)];
                xv = inb ? xv : 0.f;
                float nv = (v[i] + xv) * 0.5f;
                float sp = (nv >= 1.0f) ? 1.0f : 0.0f;
                v[i] = nv * (1.0f - sp);
                sLDS[(e < SN) ? e : SN] = inb ? sp : 0.f;
            }
        }
        __syncthreads();

        // ---- conv1 via WMMA: 21 groups of 16 h-pixels, K=9 in 3 x K=4 ----
        // Incremental per-lane state (no divisions in the loop):
        int p   = n;                      // A-row pixel index (row-major 18x18)
        int hx  = n;                      // p % 18  (n < 18)
        int abw = n;                      // hy*SDIM + hx  (hy = 0)
        int p0  = (lane >> 4) << 3;       // store base pixel (D rows)
        int sb  = p0 * 4 + n;             // store word offset pp*4 + n
        int hx2 = p0, hy2 = 0;            // border-only: store pixel coords
        for (int g = 0; g < 21; ++g) {
            bool pin = (p < HN);
            v8f acc = {0.f, 0.f, 0.f, 0.f, 0.f, 0.f, 0.f, 0.f};
#pragma unroll
            for (int kb = 0; kb < 3; ++kb) {
                v2f a;
                float v0 = sLDS[abw + toff[kb * 2 + 0]];
                float v1 = sLDS[abw + toff[kb * 2 + 1]];
                a.x = (kval[kb * 2 + 0] && pin) ? v0 : 0.f;
                a.y = (kval[kb * 2 + 1] && pin) ? v1 : 0.f;
                acc = __builtin_amdgcn_wmma_f32_16x16x4_f32(
                    false, a, false, b1[kb], (short)0, acc, false, false);
            }
            // Branchless h store; invalid lanes hit the dump word.
#pragma unroll
            for (int j = 0; j < 8; ++j) {
                int  pp    = p0 + j;
                bool valid = (n < 4) && (pp < HN);
                float hv = acc[j] > 0.f ? acc[j] : 0.f;       // ReLU
                if (!INTERIOR) {                               // zero-pad h at border
                    int  hxj = hx2 + j;
                    bool w   = (hxj >= HDIM);
                    int  hyj = hy2 + (w ? 1 : 0);
                    hxj      = w ? hxj - HDIM : hxj;
                    int gy = oy0 + hyj - 1, gx = ox0 + hxj - 1;
                    bool inimg = (gy >= 0) && (gy < IMG) && (gx >= 0) && (gx < IMG);
                    hv = inimg ? hv : 0.f;
                }
                int addr = valid ? (sb + j * 4) : (HN * 4);   // dump word
                hLDS[addr] = hv;
            }
            // advance group state
            p += 16; p0 += 16; sb += 64;
            bool wrap = (hx + 16 >= HDIM);
            hx  = hx + 16 - (wrap ? HDIM : 0);
            abw += wrap ? (16 + (SDIM - HDIM)) : 16;          // +18 on wrap, else +16
            if (!INTERIOR) {
                bool w2 = (hx2 + 16 >= HDIM);
                hx2 = hx2 + 16 - (w2 ? HDIM : 0);
                hy2 += w2 ? 1 : 0;
            }
        }
        __syncthreads();

        // ---- conv2 via WMMA: per output row g, 9 taps, K=4 channels each ----
        int gofs = 0;
        for (int g = 0; g < 16; ++g) {
            v8f acc = {0.f, 0.f, 0.f, 0.f, 0.f, 0.f, 0.f, 0.f};
#pragma unroll
            for (int tap = 0; tap < 9; ++tap) {
                v2f a;
                a.x = hLDS[gofs + c2[tap]];
                a.y = hLDS[gofs + c2[tap] + 1];
                acc = __builtin_amdgcn_wmma_f32_16x16x4_f32(
                    false, a, false, b2[tap], (short)0, acc, false, false);
            }
            // y in column 0 -> lanes 0 (pixels 0..7) and 16 (pixels 8..15)
            if (n == 0) {
                int cbase = (lane >> 4) << 3;
#pragma unroll
                for (int j = 0; j < 8; ++j) {
                    float yv = acc[j] > 0.f ? acc[j] : 0.f;    // ReLU
                    yacc[g * 16 + cbase + j] += yv;
                }
            }
            gofs += HDIM * 4;
        }
        __syncthreads();
    }

    // ---- time mean -> global ----
#pragma unroll
    for (int i = 0; i < 8; ++i) {
        int p = i * 32 + lane;
        int r = p >> 4, c = p & 15;
        out[(size_t)(oy0 + r) * IMG + (ox0 + c)] = yacc[p] * (1.0f / (float)T_STEPS);
    }
}

__global__ __launch_bounds__(32)
void lif_snn_wmma_kernel(const float* __restrict__ x_seq,
                         const float* __restrict__ w1,   // [4][1][3][3]
                         const float* __restrict__ w2,   // [1][4][3][3]
                         float* __restrict__ out)        // [512][512]
{
    __shared__ float sLDS[SPAD];              // spikes 20x20 (+dump/pad)
    __shared__ float hLDS[HN * 4 + 4];        // conv1 out 18x18 x 4ch (+dump word)
    __shared__ float yacc[TILE * TILE];       // time accumulator
    __shared__ float xbuf[2][SN];             // TDM double buffer for x tiles

    const int lane = threadIdx.x;
    const int bx = blockIdx.x, by = blockIdx.y;
    const int ox0 = bx * TILE, oy0 = by * TILE;
    const bool interior = (bx > 0) && (bx < IMG / TILE - 1) &&
                          (by > 0) && (by < IMG / TILE - 1);

    if (interior)
        run_tile<true>(x_seq, w1, w2, out, sLDS, hLDS, yacc, xbuf, lane, ox0, oy0);
    else
        run_tile<false>(x_seq, w1, w2, out, sLDS, hLDS, yacc, xbuf, lane, ox0, oy0);
}

extern "C" void kernel_launch(void* const* d_in, const int* in_sizes, int n_in,
                              void* d_out, int out_size, void* d_ws, size_t ws_size,
                              hipStream_t stream) {
    const float* x_seq = (const float*)d_in[0];  // [96,512,512]
    const float* w1    = (const float*)d_in[1];  // [4,1,3,3]
    const float* w2    = (const float*)d_in[2];  // [1,4,3,3]
    float* out = (float*)d_out;                  // [512,512]
    (void)in_sizes; (void)n_in; (void)out_size; (void)d_ws; (void)ws_size;

    dim3 grid(IMG / TILE, IMG / TILE);   // 32 x 32 tiles
    dim3 block(32, 1, 1);                // one wave32 per tile
    lif_snn_wmma_kernel<<<grid, block, 0, stream>>>(x_seq, w1, w2, out);
}